// Model_17575006175800
// MI455X (gfx1250) — compile-verified
//
#include <hip/hip_runtime.h>

typedef float v2f __attribute__((ext_vector_type(2)));
typedef float v4f __attribute__((ext_vector_type(4)));
typedef float v8f __attribute__((ext_vector_type(8)));

#define N_EXPERTS 48
#define D_IN      163840
#define THREADS   1024
#define NWAVES    (THREADS / 32)

// out[b,k] = sum_d x[b,d] * W[e(b),d,k] + bias[e(b),k],  e(b) = (t[b]-1) mod 48
__global__ __launch_bounds__(THREADS) void moe_expert_dot(
    const float* __restrict__ x,     // [B, D_IN] (flattened [B,2,1280,8,8])
    const int*   __restrict__ t,     // [B]
    const float* __restrict__ W,     // [48, D_IN, 2]
    const float* __restrict__ bias,  // [48, 2]
    float*       __restrict__ out)   // [B, 2]
{
    const int b    = blockIdx.x;
    const int tid  = threadIdx.x;
    const int lane = tid & 31;
    const int wid  = tid >> 5;

    const int e = (t[b] + (N_EXPERTS - 1)) % N_EXPERTS;  // torch t-1 with wrap

    const v4f* __restrict__ x4 = reinterpret_cast<const v4f*>(x + (size_t)b * D_IN);
    const v4f* __restrict__ w4 = reinterpret_cast<const v4f*>(W + (size_t)e * (2 * D_IN));

    float acc0 = 0.f, acc1 = 0.f;
    constexpr int kIters = D_IN / 4 / THREADS;  // 40
#pragma unroll 4
    for (int it = 0; it < kIters; ++it) {
        const int d4 = it * THREADS + tid;
        // x is read exactly once across the whole dispatch: stream it (NT).
        v4f xv = __builtin_nontemporal_load(x4 + d4);
        // W working set (60 MB) fits in 192 MB L2: keep temporal.
        v4f w0 = w4[2 * d4 + 0];   // (d,k): d*2+k -> two b128 per x b128
        v4f w1 = w4[2 * d4 + 1];
        acc0 = fmaf(xv.x, w0.x, acc0); acc1 = fmaf(xv.x, w0.y, acc1);
        acc0 = fmaf(xv.y, w0.z, acc0); acc1 = fmaf(xv.y, w0.w, acc1);
        acc0 = fmaf(xv.z, w1.x, acc0); acc1 = fmaf(xv.z, w1.y, acc1);
        acc0 = fmaf(xv.w, w1.z, acc0); acc1 = fmaf(xv.w, w1.w, acc1);
    }

    // ---- exact wave-level reduction via V_WMMA_F32_16X16X4_F32 ----
    // A (16x4 f32): lane L -> row L%16; VGPR{0,1} = K{0,1} (L<16) or K{2,3} (L>=16).
    v2f A; A.x = acc0; A.y = acc1;
    // B (4x16): col0 selects K=0,2 (acc0), col1 selects K=1,3 (acc1).
    // B VGPR0: lanes0-15 = row K=0, lanes16-31 = row K=2; VGPR1: K=1 / K=3.
    v2f Bm;
    Bm.x = (lane == 0 || lane == 16) ? 1.f : 0.f;
    Bm.y = (lane == 1 || lane == 17) ? 1.f : 0.f;
    v8f Cz = {0.f, 0.f, 0.f, 0.f, 0.f, 0.f, 0.f, 0.f};
    // D[m,0] = acc0(m)+acc0(m+16); D[m,1] = acc1(m)+acc1(m+16)
    v8f Dm = __builtin_amdgcn_wmma_f32_16x16x4_f32(
        /*neg_a=*/false, A, /*neg_b=*/false, Bm,
        /*c_mod=*/(short)0, Cz, /*reuse_a=*/false, /*reuse_b=*/false);

    // Sum D over its 8 VGPRs: lane n (n<16) now holds sum_{m=0..7} D[m,n],
    // lane 16+n holds sum_{m=8..15} D[m,n].
    float s = Dm[0] + Dm[1] + Dm[2] + Dm[3] + Dm[4] + Dm[5] + Dm[6] + Dm[7];

    float tot0 = __int_as_float(__builtin_amdgcn_readlane(__float_as_int(s), 0)) +
                 __int_as_float(__builtin_amdgcn_readlane(__float_as_int(s), 16));
    float tot1 = __int_as_float(__builtin_amdgcn_readlane(__float_as_int(s), 1)) +
                 __int_as_float(__builtin_amdgcn_readlane(__float_as_int(s), 17));

    __shared__ float red[NWAVES][2];
    if (lane == 0) { red[wid][0] = tot0; red[wid][1] = tot1; }
    __syncthreads();

    if (tid == 0) {
        float o0 = bias[2 * e + 0];
        float o1 = bias[2 * e + 1];
#pragma unroll
        for (int w = 0; w < NWAVES; ++w) { o0 += red[w][0]; o1 += red[w][1]; }
        out[2 * b + 0] = o0;
        out[2 * b + 1] = o1;
    }
}

extern "C" void kernel_launch(void* const* d_in, const int* in_sizes, int n_in,
                              void* d_out, int out_size, void* d_ws, size_t ws_size,
                              hipStream_t stream) {
    const float* x    = (const float*)d_in[0];
    const int*   t    = (const int*)d_in[1];
    const float* W    = (const float*)d_in[2];
    const float* bias = (const float*)d_in[3];
    float*       out  = (float*)d_out;

    const int B = in_sizes[0] / D_IN;  // 512
    moe_expert_dot<<<B, THREADS, 0, stream>>>(x, t, W, bias, out);
}